// WeightLocalModule_35871566856221
// MI455X (gfx1250) — compile-verified
//
#include <hip/hip_runtime.h>

// out[n, d] = x[n, d] * w[d]   (N = 200000, D = 512, fp32)
// Pure streaming kernel: 128-bit non-temporal load/store, loop-invariant
// weight chunk per thread, gfx1250 global_prefetch ahead of the stream.

typedef float v4f __attribute__((ext_vector_type(4)));

#ifndef BLOCK_THREADS
#define BLOCK_THREADS 256   // 8 wave32s per block
#endif

__global__ __launch_bounds__(BLOCK_THREADS) void wl_colscale_kernel(
    const v4f* __restrict__ x4,
    const float* __restrict__ w,
    v4f* __restrict__ o4,
    long long n4)   // total float4 count = N*D/4
{
    const long long stride = (long long)gridDim.x * (long long)blockDim.x;
    long long idx = (long long)blockIdx.x * blockDim.x + threadIdx.x;

    // D = 512 floats = 128 float4 groups per row. stride is a multiple of 128
    // (blockDim=256), so this thread's column group is loop-invariant.
    const int colg = (int)(idx & 127);
    const v4f w4 = ((const v4f*)w)[colg];   // temporal load; 2KB of w stays hot

    for (; idx < n4; idx += stride) {
        // Prefetch ~2 grid-strides ahead (speculative: OOB is silently dropped)
        __builtin_prefetch((const void*)&x4[idx + 2 * stride], 0, 1);

        v4f v = __builtin_nontemporal_load(&x4[idx]);   // global_load_b128, NT
        v4f r = v * w4;
        __builtin_nontemporal_store(r, &o4[idx]);       // global_store_b128, NT
    }
}

extern "C" void kernel_launch(void* const* d_in, const int* in_sizes, int n_in,
                              void* d_out, int out_size, void* d_ws, size_t ws_size,
                              hipStream_t stream) {
    const float* x = (const float*)d_in[0];        // [N, D] fp32
    const float* w = (const float*)d_in[1];        // [D, 1] fp32
    float*       o = (float*)d_out;                // [N, D] fp32

    const long long total = (long long)in_sizes[0];    // N*D (multiple of 4)
    const long long n4    = total >> 2;

    // 8192 blocks x 256 threads = 64K wave32s in flight: enough outstanding
    // b128 loads to cover HBM latency at 23.3 TB/s; grid-stride handles rest.
    const int blocks = 8192;

    wl_colscale_kernel<<<dim3(blocks), dim3(BLOCK_THREADS), 0, stream>>>(
        (const v4f*)x, w, (v4f*)o, n4);
}